// GraphConvolutionLayer_13176959664541
// MI455X (gfx1250) — compile-verified
//
#include <hip/hip_runtime.h>
#include <hip/hip_bf16.h>

#define N_NODES 100000
#define N_EDGES 1600000
#define FEATS   64
#define LDSP    66   // LDS row pitch in floats (even -> 8B aligned v2f, stride-2 banks)

typedef __attribute__((ext_vector_type(2))) float v2f;
typedef __attribute__((ext_vector_type(8))) float v8f;

// ---------------- Kernel 1: zero the aggregation buffer (d_out doubles as agg) ----
__global__ __launch_bounds__(256) void gcn_zero_kernel(float4* __restrict__ agg4) {
    int i = blockIdx.x * 256 + threadIdx.x;
    agg4[i] = make_float4(0.f, 0.f, 0.f, 0.f);
}

// ---------------- Kernel 2: edge scatter-add (segment sum) ------------------------
// One thread per (edge, 4-feature chunk): coalesced float4 gather of features[src],
// 4 consecutive global_atomic_add_f32 into agg[dst]. Both arrays are L2-resident.
__global__ __launch_bounds__(256) void gcn_scatter_kernel(const float* __restrict__ feats,
                                                          const int*   __restrict__ src,
                                                          const int*   __restrict__ dst,
                                                          float*       __restrict__ agg) {
    int gtid = blockIdx.x * 256 + threadIdx.x;   // 25.6M threads total
    int e  = gtid >> 4;                          // 16 threads per edge
    int f0 = (gtid & 15) << 2;                   // 4 features per thread
    int s = src[e];
    int d = dst[e];
    const float4 v = *(const float4*)(feats + s * FEATS + f0);
    float* p = agg + d * FEATS + f0;
    atomicAdd(p + 0, v.x);
    atomicAdd(p + 1, v.y);
    atomicAdd(p + 2, v.z);
    atomicAdd(p + 3, v.w);
}

// ---------------- Kernel 3: in-place f32 WMMA GEMM + bias + ReLU ------------------
// h[m, n] = relu( sum_k agg[m, k] * W[n, k] + b[n] ),  W is [out=64, in=64].
// Each wave owns a 16-row tile: A tile (16x64) fully loaded to registers first,
// so overwriting agg in place is safe.
__global__ __launch_bounds__(64) void gcn_gemm_relu_kernel(float*       __restrict__ agg_out,
                                                           const float* __restrict__ W,
                                                           const float* __restrict__ bias) {
    __shared__ float Wl[64 * LDSP];

    const int t = threadIdx.x;
    // Cooperative, coalesced load of W into padded LDS: Wl[n*LDSP + k] = W[n*64 + k]
    #pragma unroll
    for (int i = 0; i < 64; ++i) {
        int idx = i * 64 + t;
        Wl[(idx >> 6) * LDSP + (idx & 63)] = W[idx];
    }
    __syncthreads();

    const int wave = t >> 5;
    const int lane = t & 31;
    const int half = lane >> 4;   // 0: lanes 0-15, 1: lanes 16-31
    const int l16  = lane & 15;
    const int m0   = (blockIdx.x * 2 + wave) * 16;   // this wave's row tile

    // A-frag per ISA layout for V_WMMA_F32_16X16X4_F32:
    //   lane L holds row (L&15); VGPR pair kt holds K = 4*kt + 2*half + {0,1}
    v2f a[16];
    const float* arow = agg_out + (m0 + l16) * FEATS + 2 * half;
    #pragma unroll
    for (int kt = 0; kt < 16; ++kt)
        a[kt] = *(const v2f*)(arow + 4 * kt);

    #pragma unroll
    for (int nt = 0; nt < 4; ++nt) {
        const int n = nt * 16 + l16;             // output column this lane owns
        const float bv = bias[n];
        v8f acc = { bv, bv, bv, bv, bv, bv, bv, bv };  // bias folded into C

        // B-frag (4x16): lane L holds col (L&15); VGPR pair = K = 4*kt + 2*half + {0,1}
        // B[k][n] = W[n][k]  (W stored row-major [out, in])
        const float* brow = &Wl[n * LDSP + 2 * half];
        #pragma unroll
        for (int kt = 0; kt < 16; ++kt) {
            v2f bf = *(const v2f*)(brow + 4 * kt);
            acc = __builtin_amdgcn_wmma_f32_16x16x4_f32(
                /*neg_a=*/false, a[kt], /*neg_b=*/false, bf,
                /*c_mod=*/(short)0, acc, /*reuse_a=*/false, /*reuse_b=*/false);
        }

        // D layout: VGPR v -> row m0 + v + 8*half, col n. ReLU + in-place store.
        #pragma unroll
        for (int v = 0; v < 8; ++v) {
            float x = acc[v];
            agg_out[(m0 + v + 8 * half) * FEATS + n] = x > 0.f ? x : 0.f;
        }
    }
}

extern "C" void kernel_launch(void* const* d_in, const int* in_sizes, int n_in,
                              void* d_out, int out_size, void* d_ws, size_t ws_size,
                              hipStream_t stream) {
    const float* features = (const float*)d_in[0];   // [100000, 64] f32
    const int*   src      = (const int*)  d_in[1];   // [1600000] i32
    const int*   dst      = (const int*)  d_in[2];   // [1600000] i32
    const float* W        = (const float*)d_in[3];   // [64, 64] f32
    const float* bias     = (const float*)d_in[4];   // [64] f32
    float*       out      = (float*)d_out;           // [100000, 64] f32 (also agg buffer)

    // 1) zero agg (= d_out): 6.4M floats as float4 -> 1.6M threads
    gcn_zero_kernel<<<(N_NODES * FEATS / 4) / 256, 256, 0, stream>>>((float4*)out);

    // 2) scatter-add: 1.6M edges * 16 threads = 25.6M threads
    gcn_scatter_kernel<<<(N_EDGES * 16) / 256, 256, 0, stream>>>(features, src, dst, out);

    // 3) in-place WMMA GEMM + bias + ReLU: 3125 blocks * 2 waves * 16 rows = 100000
    gcn_gemm_relu_kernel<<<N_NODES / 32, 64, 0, stream>>>(out, W, bias);
}